// Pna_67448166416674
// MI455X (gfx1250) — compile-verified
//
#include <hip/hip_runtime.h>
#include <math.h>

// CDNA5 (gfx1250) wave32 fp32 WMMA implementation of the PNA reference.
// B=16, N=256, D_IN=256, D_H=64, D_MID=128. Dense GEMMs use
// V_WMMA_F32_16X16X4_F32 with 32x32 register blocking (4 accumulators/wave).
// B-operand addressing is split into a wave-uniform row pointer (SGPR, SADDR
// loads) plus loop-invariant per-lane offsets, so the unrolled inner loop has
// no per-lane address arithmetic and minimal VGPR pressure.
// Sparse aggregation uses compacted u8 neighbor lists (~14 of 256) in LDS.

typedef __attribute__((ext_vector_type(2))) float v2f;
typedef __attribute__((ext_vector_type(8))) float v8f;

#define NEG_FILL 1e30f

static __device__ __forceinline__ v8f wmma4(v2f a, v2f b, v8f c) {
    // 8 args: (neg_a, A, neg_b, B, c_mod, C, reuse_a, reuse_b)
    return __builtin_amdgcn_wmma_f32_16x16x4_f32(false, a, false, b,
                                                 (short)0, c, false, false);
}

// ---------------------------------------------------------------------------
// Kernel 1: degrees + amp/att scalers + compacted neighbor lists.
// One block per graph b; N(=256) threads, thread i owns adjacency row i.
// ---------------------------------------------------------------------------
__global__ void pna_scalers(const unsigned char* __restrict__ adj,
                            float* __restrict__ deg,
                            float* __restrict__ amp,
                            float* __restrict__ att,
                            unsigned char* __restrict__ nbr,
                            int* __restrict__ cnt,
                            int N) {
    int b = blockIdx.x;
    int i = threadIdx.x;
    int bi = b * N + i;
    const unsigned char* row = adj + (size_t)bi * N;
    unsigned char* nl = nbr + (size_t)bi * N;

    int c = 0;
    for (int j = 0; j < N; ++j)
        if (row[j] || j == i) nl[c++] = (unsigned char)j;   // adjacency + self-loop
    cnt[bi] = c;

    float d = (float)c;
    float ld = logf(d + 1.f);

    __shared__ float red[256];
    red[i] = ld;
    __syncthreads();
    for (int s = 128; s > 0; s >>= 1) {
        if (i < s) red[i] += red[i + s];
        __syncthreads();
    }
    float delta = red[0] / (float)N;   // per-graph mean log-degree

    deg[bi] = d;
    amp[bi] = ld / delta;
    att[bi] = delta / ld;
}

// ---------------------------------------------------------------------------
// Kernel 2: C[M,Nd] = A[M,K] @ B[K,Nd] + bias[Nd], optional ReLU.
// fp32 WMMA 16x16x4; one wave computes a 32x32 C tile (4 accumulators),
// reusing each A/B fragment twice. Requires M%32==0, Nd%32==0, K%64==0.
// VGPR layouts per CDNA5 ISA 7.12.2:
//   A (16x4): lanes<16 -> K={k,k+1}; lanes>=16 -> K={k+2,k+3}; M = lane&15.
//   B (4x16): same K split across lane halves; N = lane&15.
//   C (16x16): VGPR r -> row r (lanes 0-15) / row r+8 (lanes 16-31).
// Tile index is wave-uniform -> no divergence, EXEC stays all-1s for WMMA.
// Outer loop steps K by 64 (one prefetch per body); inner 16 steps fully
// unrolled. A loads: global_load_b64 at immediate offsets. B loads: uniform
// SGPR row pointer (bumped by 4*Nd per step) + invariant per-lane offsets.
// ---------------------------------------------------------------------------
__global__ void gemm_bias_wmma32(const float* __restrict__ A,
                                 const float* __restrict__ B,
                                 const float* __restrict__ bias,
                                 float* __restrict__ C,
                                 int M, int K, int Nd, int relu) {
    int lane = threadIdx.x & 31;
    int wave = threadIdx.x >> 5;

    int tiles_n = Nd >> 5;
    int tile = blockIdx.x * (blockDim.x >> 5) + wave;   // wave-uniform
    int tm = tile / tiles_n;
    int tn = tile - tm * tiles_n;
    int m0 = tm << 5;
    int n0 = tn << 5;

    int half = lane >> 4;          // 0: lanes 0-15, 1: lanes 16-31
    int koff = half << 1;          // K sub-offset 0 or 2
    int l15  = lane & 15;

    // A fragment bases (contiguous aligned float pairs -> global_load_b64)
    const float* ap0 = A + (size_t)(m0 + l15) * K + koff;
    const float* ap1 = ap0 + (size_t)16 * K;

    // B: loop-invariant lane offsets; uniform row pointer advances per step.
    int nc0       = n0 + l15;
    int lane_off  = koff * Nd + nc0;   // element (row koff, col nc0)
    int lane_off2 = lane_off + Nd;     // element (row koff+1, col nc0)

    v8f c00 = {}, c01 = {}, c10 = {}, c11 = {};

    const float* brow = B;             // wave-uniform; += 4*Nd per K-step
    for (int k0 = 0; k0 < K; k0 += 64) {
        if (k0 + 64 < K) {
            // speculative read-ahead on the A streams -> global_prefetch_b8
            __builtin_prefetch(ap0 + 64, 0, 3);
            __builtin_prefetch(ap1 + 64, 0, 3);
        }
        #pragma unroll
        for (int u = 0; u < 16; ++u) {
            v2f a0 = *reinterpret_cast<const v2f*>(ap0 + u * 4);
            v2f a1 = *reinterpret_cast<const v2f*>(ap1 + u * 4);
            v2f b0, b1;
            b0.x = brow[lane_off];        // col nc0, rows kk / kk+1
            b0.y = brow[lane_off2];
            b1.x = brow[lane_off + 16];   // col nc0+16 (immediate offset:64)
            b1.y = brow[lane_off2 + 16];
            c00 = wmma4(a0, b0, c00);
            c01 = wmma4(a0, b1, c01);
            c10 = wmma4(a1, b0, c10);
            c11 = wmma4(a1, b1, c11);
            brow += (size_t)(Nd << 2);    // uniform SALU bump
        }
        ap0 += 64;
        ap1 += 64;
    }

    float bv0 = bias[nc0];
    float bv1 = bias[nc0 + 16];
    int r0 = m0 + (half << 3);          // rows for c0x
    int r1 = r0 + 16;                   // rows for c1x
    float* crow0 = C + (size_t)r0 * Nd + nc0;
    float* crow1 = C + (size_t)r1 * Nd + nc0;
    #pragma unroll
    for (int r = 0; r < 8; ++r) {
        float v00 = c00[r] + bv0, v01 = c01[r] + bv1;
        float v10 = c10[r] + bv0, v11 = c11[r] + bv1;
        if (relu) {
            v00 = fmaxf(v00, 0.f); v01 = fmaxf(v01, 0.f);
            v10 = fmaxf(v10, 0.f); v11 = fmaxf(v11, 0.f);
        }
        crow0[0]  = v00;
        crow0[16] = v01;
        crow1[0]  = v10;
        crow1[16] = v11;
        crow0 += Nd;
        crow1 += Nd;
    }
}

// ---------------------------------------------------------------------------
// Kernel 3: fused PNA aggregation + scaler expansion (sparse).
// One block per destination node (b,i); D threads, thread d owns feature d.
// Compacted neighbor list cached in LDS; loop trip count = degree (~14),
// uniform across the block -> no divergence.
// Emits scaled[B,N,12D] = [mean,max,min,std | *amp | *att].
// ---------------------------------------------------------------------------
template <int D>
__global__ void pna_agg(const float* __restrict__ h,
                        const unsigned char* __restrict__ nbr,
                        const int* __restrict__ cnt,
                        const float* __restrict__ deg,
                        const float* __restrict__ amp,
                        const float* __restrict__ att,
                        float* __restrict__ scaled,
                        int N) {
    int bi = blockIdx.x;           // b*N + i
    int b  = bi >> 8;              // N == 256
    int d  = threadIdx.x;

    int cn = cnt[bi];              // block-uniform (>=1: self-loop)

    __shared__ unsigned char s_nl[256];
    const unsigned char* nl = nbr + (size_t)bi * N;
    for (int t = d; t < cn; t += D) s_nl[t] = nl[t];
    __syncthreads();

    const float* hb = h + (size_t)b * N * D;
    float s = 0.f, s2 = 0.f, mx = -NEG_FILL, mn = NEG_FILL;
    for (int c = 0; c < cn; ++c) {
        int j = s_nl[c];
        float v = hb[(size_t)j * D + d];
        s  += v;
        s2 += v * v;
        mx  = fmaxf(mx, v);
        mn  = fminf(mn, v);
    }

    float dg   = deg[bi];
    float mean = s / dg;
    float var  = s2 / dg - mean * mean;
    float sd   = sqrtf(fmaxf(var, 0.f) + 1e-5f);
    float am   = amp[bi];
    float at   = att[bi];

    float* o = scaled + (size_t)bi * 12 * D;
    o[0 * D + d] = mean;       o[1 * D + d] = mx;
    o[2 * D + d] = mn;         o[3 * D + d] = sd;
    o[4 * D + d] = mean * am;  o[5 * D + d] = mx * am;
    o[6 * D + d] = mn * am;    o[7 * D + d] = sd * am;
    o[8 * D + d] = mean * at;  o[9 * D + d] = mx * at;
    o[10 * D + d] = mn * at;   o[11 * D + d] = sd * at;
}

// ---------------------------------------------------------------------------
// Kernel 4: final prediction head: out[m] = h[m,:] . Wp + bp. One wave per row.
// ---------------------------------------------------------------------------
__global__ void pna_pred(const float* __restrict__ h,
                         const float* __restrict__ Wp,
                         const float* __restrict__ bp,
                         float* __restrict__ out,
                         int M, int D) {
    int lane = threadIdx.x & 31;
    int wave = threadIdx.x >> 5;
    int row  = blockIdx.x * (blockDim.x >> 5) + wave;
    if (row >= M) return;
    const float* hr = h + (size_t)row * D;
    float s = 0.f;
    for (int d = lane; d < D; d += 32) s += hr[d] * Wp[d];
    #pragma unroll
    for (int off = 16; off > 0; off >>= 1) s += __shfl_xor(s, off, 32);
    if (lane == 0) out[row] = s + bp[0];
}

// ---------------------------------------------------------------------------
// Orchestration. Workspace carve-out (~31 MB):
//   deg/amp/att : 3 * 4096 f32
//   cnt         : 4096 i32
//   nbr         : 4096*256 u8 (compacted neighbor indices)
//   hA, hB      : 2 * 4096*128 f32 (hA holds h0 then h2; hB holds h1)
//   scaled      : 4096*1536 f32 (reused by both layers)
// ---------------------------------------------------------------------------
extern "C" void kernel_launch(void* const* d_in, const int* in_sizes, int n_in,
                              void* d_out, int out_size, void* d_ws, size_t ws_size,
                              hipStream_t stream) {
    const float*         x     = (const float*)d_in[0];
    const unsigned char* adj   = (const unsigned char*)d_in[1];  // jax bool -> 1 byte
    const float*         Wproj = (const float*)d_in[2];
    const float*         bproj = (const float*)d_in[3];
    const float*         W1    = (const float*)d_in[4];
    const float*         b1    = (const float*)d_in[5];
    const float*         W2    = (const float*)d_in[6];
    const float*         b2    = (const float*)d_in[7];
    const float*         Wp    = (const float*)d_in[8];
    const float*         bp    = (const float*)d_in[9];
    float*               out   = (float*)d_out;

    const int B = 16, N = 256, DIN = 256, DH = 64, DMID = 128;
    const int M = B * N;  // 4096

    float*         ws  = (float*)d_ws;
    float*         deg = ws;
    float*         amp = deg + M;
    float*         att = amp + M;
    int*           cnt = (int*)(att + M);
    unsigned char* nbr = (unsigned char*)(cnt + M);              // M*N bytes
    float*         hA  = (float*)(nbr + (size_t)M * N);          // M*DMID capacity
    float*         hB  = hA + (size_t)M * DMID;
    float*         scl = hB + (size_t)M * DMID;                  // M * 12*DMID

    // 1) degree / amp / att scalers + neighbor-list compaction
    pna_scalers<<<B, N, 0, stream>>>(adj, deg, amp, att, nbr, cnt, N);

    // 2) h0 = x @ W_proj + b_proj   [4096,256]@[256,64]
    {
        int tiles = (M / 32) * (DH / 32);                        // 256
        gemm_bias_wmma32<<<tiles / 4, 128, 0, stream>>>(x, Wproj, bproj, hA,
                                                        M, DIN, DH, 0);
    }

    // 3) layer-1 aggregation -> scaled [4096, 768]
    pna_agg<DH><<<M, DH, 0, stream>>>(hA, nbr, cnt, deg, amp, att, scl, N);

    // 4) h1 = relu(scaled @ W1 + b1)  [4096,768]@[768,128]
    {
        int tiles = (M / 32) * (DMID / 32);                      // 512
        gemm_bias_wmma32<<<tiles / 4, 128, 0, stream>>>(scl, W1, b1, hB,
                                                        M, 12 * DH, DMID, 1);
    }

    // 5) layer-2 aggregation -> scaled [4096, 1536]
    pna_agg<DMID><<<M, DMID, 0, stream>>>(hB, nbr, cnt, deg, amp, att, scl, N);

    // 6) h2 = scaled @ W2 + b2   [4096,1536]@[1536,64]
    {
        int tiles = (M / 32) * (DH / 32);                        // 256
        gemm_bias_wmma32<<<tiles / 4, 128, 0, stream>>>(scl, W2, b2, hA,
                                                        M, 12 * DMID, DH, 0);
    }

    // 7) out = h2 @ W_pred + b_pred   [4096,64]@[64,1]
    pna_pred<<<(M + 7) / 8, 256, 0, stream>>>(hA, Wp, bp, out, M, DH);
}